// GroupQuantizedLinear_86371792322991
// MI455X (gfx1250) — compile-verified
//
#include <hip/hip_runtime.h>
#include <stddef.h>

// ---------------------------------------------------------------------------
// Group-quantized GEMV for MI455X (gfx1250).
//
// Roofline: 256MB of f32 weights streamed once vs 134 MFLOP => HBM-bound,
// floor ~= 268MB / 23.3TB/s ~= 11.5us. Design:
//   * every weight byte loaded exactly once, per-lane global_load_b64 with a
//     non-temporal hint (256MB > 192MB L2, zero reuse)
//   * exact f32 math via V_WMMA_F32_16X16X4_F32 (quantized weights are
//     integers in [-8,7]; accumulation stays f32 like the reference)
//   * K split 16 ways => 512 row-tiles * 16 = 8192 wave32s in flight to cover
//     HBM latency; deterministic two-stage reduction through d_ws
//   * all shapes are reference-constant => constexpr, so the inner loop is
//     just loads + med3/rndne/mul + wmma (no runtime div/mod SALU bloat).
// ---------------------------------------------------------------------------

typedef float v2f __attribute__((ext_vector_type(2)));
typedef float v8f __attribute__((ext_vector_type(8)));

constexpr int IN_DIM  = 8192;
constexpr int OUT_DIM = 8192;
constexpr int NGROUPS = 64;
constexpr int GSZ     = IN_DIM / NGROUPS;   // 128 channels per scale group

template <int KLOG>
__global__ __launch_bounds__(256) void gql_gemv_wmma_kernel(
    const float* __restrict__ W,       // [OUT_DIM, IN_DIM]
    const float* __restrict__ x,       // [IN_DIM]
    const float* __restrict__ S,       // [OUT_DIM, NGROUPS]
    float* __restrict__ part)          // [KSPLIT, OUT_DIM] (== out if KSPLIT==1)
{
    constexpr int KSPLIT = 1 << KLOG;
    constexpr int KPER   = IN_DIM / KSPLIT;   // K range per wave
    constexpr int GPER   = NGROUPS / KSPLIT;  // scale groups per wave
    constexpr int ITERS  = GSZ / 4;           // WMMAs (K=4) per group = 32

    const int lane   = threadIdx.x & 31;
    const int wave   = blockIdx.x * 8 + (threadIdx.x >> 5);
    const int kseg   = wave & (KSPLIT - 1);
    const int rowblk = wave >> KLOG;          // 16 output rows per wave
    const int h      = lane >> 4;             // lane-half: K-pair select
    const int r      = lane & 15;             // row within 16-row tile
    const int row    = rowblk * 16 + r;
    const int k0     = kseg * KPER;

    // A-matrix 16x4 f32 layout: lane L<16 holds row L {K,K+1} in {v0,v1};
    // lane L+16 holds row L {K+2,K+3}. => one aligned b64 per lane per WMMA.
    const float* wp = W + (size_t)row * IN_DIM + (size_t)(k0 + 2 * h);
    const float* xp = x + (size_t)(k0 + 2 * h);
    const float* sp = S + (size_t)row * NGROUPS + (size_t)(kseg * GPER);

    v8f acc = {};  // 16x16 f32 C/D; all 16 columns identical (B broadcast)

    for (int g = 0; g < GPER; ++g) {
        const float sc = sp[g];               // one scalar load per 32 WMMAs
#pragma unroll 8
        for (int i = 0; i < ITERS; ++i) {
            // stream weights non-temporally (no reuse; keep L2 for x/scales)
            unsigned long long wb =
                __builtin_nontemporal_load((const unsigned long long*)wp);
            float w0 = __builtin_bit_cast(float, (unsigned int)(wb & 0xffffffffULL));
            float w1 = __builtin_bit_cast(float, (unsigned int)(wb >> 32));
            float2 xv = *(const float2*)xp;   // 32KB vector, L0/L2 resident

            v2f a, b;
            // fake-quant + dequant: v_med3_f32 + v_rndne_f32 + v_mul_f32
            a.x = __builtin_rintf(__builtin_amdgcn_fmed3f(w0, -8.0f, 7.0f)) * sc;
            a.y = __builtin_rintf(__builtin_amdgcn_fmed3f(w1, -8.0f, 7.0f)) * sc;
            // B 4x16: row K broadcast across all 16 columns -> same per-lane
            // {K,K+1}/{K+2,K+3} pairing as A.
            b.x = xv.x;
            b.y = xv.y;

            acc = __builtin_amdgcn_wmma_f32_16x16x4_f32(
                /*neg_a=*/false, a, /*neg_b=*/false, b,
                /*c_mod=*/(short)0, acc, /*reuse_a=*/false, /*reuse_b=*/false);

            wp += 4;
            xp += 4;
        }
    }

    // D layout: VGPR j, lanes 0-15 -> M=j ; lanes 16-31 -> M=8+j.
    // Columns identical, so lane 0 holds rows 0..7, lane 16 rows 8..15.
    if (r == 0) {
        float* p = part + (size_t)kseg * OUT_DIM + (size_t)rowblk * 16 + 8 * h;
#pragma unroll
        for (int j = 0; j < 8; ++j) p[j] = acc[j];
    }
}

__global__ __launch_bounds__(256) void gql_reduce_kernel(
    const float* __restrict__ part, float* __restrict__ out, int ksplit)
{
    int i = blockIdx.x * blockDim.x + threadIdx.x;
    if (i < OUT_DIM) {
        float s = 0.0f;
        for (int k = 0; k < ksplit; ++k) s += part[(size_t)k * OUT_DIM + i];
        out[i] = s;
    }
}

extern "C" void kernel_launch(void* const* d_in, const int* in_sizes, int n_in,
                              void* d_out, int out_size, void* d_ws, size_t ws_size,
                              hipStream_t stream) {
    const float* x = (const float*)d_in[0];          // [IN_DIM]
    const float* W = (const float*)d_in[1];          // [OUT_DIM, IN_DIM]
    const float* S = (const float*)d_in[2];          // [OUT_DIM, NGROUPS]
    float* out = (float*)d_out;
    (void)in_sizes; (void)n_in; (void)out_size;

    // K-split: want ~8K waves to cover 23.3TB/s * latency; shrink only if the
    // scratch buffer can't hold the partials (deterministic: depends only on
    // fixed problem/ws sizes).
    int klog = 4;                                    // KSPLIT = 16
    while (klog > 0 && (size_t)(1u << klog) * (size_t)OUT_DIM * sizeof(float) > ws_size)
        --klog;
    const int ksplit = 1 << klog;

    float* part = (ksplit == 1) ? out : (float*)d_ws;

    const int waves  = (OUT_DIM / 16) * ksplit;      // 16 rows per wave
    const int blocks = waves / 8;                    // 8 wave32 per 256-thr block

    switch (klog) {
        case 4: gql_gemv_wmma_kernel<4><<<blocks, 256, 0, stream>>>(W, x, S, part); break;
        case 3: gql_gemv_wmma_kernel<3><<<blocks, 256, 0, stream>>>(W, x, S, part); break;
        case 2: gql_gemv_wmma_kernel<2><<<blocks, 256, 0, stream>>>(W, x, S, part); break;
        case 1: gql_gemv_wmma_kernel<1><<<blocks, 256, 0, stream>>>(W, x, S, part); break;
        default: gql_gemv_wmma_kernel<0><<<blocks, 256, 0, stream>>>(W, x, S, part); break;
    }

    if (ksplit > 1) {
        gql_reduce_kernel<<<(OUT_DIM + 255) / 256, 256, 0, stream>>>(part, out, ksplit);
    }
}